// YoloLossV2_71734543778440
// MI455X (gfx1250) — compile-verified
//
#include <hip/hip_runtime.h>
#include <math.h>

typedef __attribute__((ext_vector_type(2))) float v2f;
typedef __attribute__((ext_vector_type(8))) float v8f;

__constant__ float c_anchors[3][3][2] = {
    {{10.f, 13.f}, {16.f, 30.f}, {33.f, 23.f}},
    {{30.f, 61.f}, {62.f, 45.f}, {59.f, 119.f}},
    {{116.f, 90.f}, {156.f, 198.f}, {373.f, 326.f}}};
__constant__ float c_len[3] = {8.f, 16.f, 32.f};

// ---------------------------------------------------------------------------
// Wave-level 32-lane sum via V_WMMA_F32_16X16X4_F32.
// A (16x4 f32): lane L<16 holds A[L][0]=v, A[L][1]=0; lane L>=16 holds
// A[L-16][2]=v, A[L-16][3]=0. With B = all-ones, D[m][n] = v[m] + v[m+16].
// Each lane sums its 8 D VGPRs (= 8 rows of its half) and one xor-16 shuffle
// adds the two halves -> full 32-lane sum in every lane. EXEC must be all 1s.
// ---------------------------------------------------------------------------
__device__ __forceinline__ float wave_sum32(float v) {
  v2f a; a[0] = v;   a[1] = 0.0f;
  v2f b; b[0] = 1.0f; b[1] = 1.0f;
  v8f c = {0.f, 0.f, 0.f, 0.f, 0.f, 0.f, 0.f, 0.f};
  v8f d = __builtin_amdgcn_wmma_f32_16x16x4_f32(
      false, a, false, b, (short)0, c, false, false);
  float s = d[0] + d[1] + d[2] + d[3] + d[4] + d[5] + d[6] + d[7];
  s += __shfl_xor(s, 16, 32);
  return s;
}

// Deterministic 256-thread block sum; valid result in thread 0.
__device__ __forceinline__ float block_sum256(float v) {
  __shared__ float red[8];
  __syncthreads();                       // safe reuse of `red`
  float s = wave_sum32(v);               // uniform: EXEC all 1s here
  if ((threadIdx.x & 31) == 0) red[threadIdx.x >> 5] = s;
  __syncthreads();
  float t = 0.f;
  if (threadIdx.x == 0) {
#pragma unroll
    for (int i = 0; i < 8; ++i) t += red[i];
  }
  return t;
}

__device__ __forceinline__ float bce0(float x) {  // BCE(x, 0)
  return fmaxf(x, 0.f) + log1pf(__expf(-fabsf(x)));
}

// ---------------------------------------------------------------------------
// Kernel 1: sum of BCE(obj_logit, 0) over ALL positions.
// Contiguous stream of prd_f (104 MB) with incremental channel tracking —
// ~4x less DRAM line traffic than a 340-byte-strided gather of channel 4.
// ---------------------------------------------------------------------------
__global__ __launch_bounds__(256) void noobj_kernel(const float* __restrict__ prd,
                                                    int n, int ch_count,
                                                    float* __restrict__ partials) {
  const int tid  = blockIdx.x * blockDim.x + threadIdx.x;
  const int T    = gridDim.x * blockDim.x;
  const int step = T % ch_count;
  int ch = tid % ch_count;
  float acc = 0.f;
  for (int f = tid; f < n; f += T) {
    float x = prd[f];
    if (ch == 4) acc += bce0(x);
    ch += step;
    if (ch >= ch_count) ch -= ch_count;
  }
  float t = block_sum256(acc);
  if (threadIdx.x == 0) partials[blockIdx.x] = t;
}

// ---------------------------------------------------------------------------
// Kernel 2: segmentation NLL:  sum over pixels of (logsumexp_c - seg[label]).
// One thread per pixel, 21 channels fully unrolled (constant register
// indexing; label value captured by select to avoid dynamic indexing).
// ---------------------------------------------------------------------------
__global__ __launch_bounds__(256) void seg_kernel(const float* __restrict__ seg,
                                                  const int* __restrict__ seglab,
                                                  int npix, int hw,
                                                  float* __restrict__ partials) {
  constexpr int SEGC = 21;
  const int i = blockIdx.x * blockDim.x + threadIdx.x;
  float term = 0.f;
  if (i < npix) {
    const int b   = i / hw;
    const int rem = i - b * hw;
    const int base = b * SEGC * hw + rem;
    const int label = seglab[i];
    float v[SEGC];
    float m  = -3.402823466e+38f;
    float vl = 0.f;
#pragma unroll
    for (int c = 0; c < SEGC; ++c) {
      v[c] = seg[base + c * hw];
      m = fmaxf(m, v[c]);
      vl = (c == label) ? v[c] : vl;
    }
    float s = 0.f;
#pragma unroll
    for (int c = 0; c < SEGC; ++c) s += __expf(v[c] - m);
    term = (m + __logf(s)) - vl;         // -log_softmax at label
  }
  float t = block_sum256(term);
  if (threadIdx.x == 0) partials[blockIdx.x] = t;
}

// ---------------------------------------------------------------------------
// Kernel 3: the 4096 gathered cells: CIoU, obj BCE, BCE(ps4,0) (for noobj
// subtraction), and 80-way log-softmax class NLL. One thread per item.
// ---------------------------------------------------------------------------
__global__ __launch_bounds__(256) void gather_kernel(
    const float* __restrict__ prd, const float* __restrict__ tgt,
    const int* __restrict__ idx, const int* __restrict__ fidx_p, int N,
    int A, int H, int W, int CH, float* __restrict__ partials, int GG) {
  const int i = blockIdx.x * blockDim.x + threadIdx.x;
  float t_ciou = 0.f, t_lobj = 0.f, t_bce0 = 0.f, t_lcls = 0.f;
  if (i < N) {
    const int f  = *fidx_p;
    const int b  = idx[i];
    const int a  = idx[N + i];
    const int h  = idx[2 * N + i];
    const int w  = idx[3 * N + i];
    const int pos = ((b * A + a) * H + h) * W + w;
    const float* ps = prd + (long)pos * CH;
    const float* ts = tgt + (long)pos * 10 + 1;   // target[...,1:5]

    const float lw = c_len[f], lh = c_len[f];
    const float aw = c_anchors[f][a][0], ah = c_anchors[f][a][1];

    const float s0 = 1.f / (1.f + __expf(-ps[0]));
    const float s1 = 1.f / (1.f + __expf(-ps[1]));
    const float px  = lw * (s0 + (float)w);
    const float py  = lh * (s1 + (float)h);
    const float pcw = aw * __expf(ps[2]);
    const float pch = ah * __expf(ps[3]);

    const float eps = 1e-7f;
    // b1 = pbox (xywh), b2 = ts interpreted as xywh (as the reference does)
    const float b1x1 = px - pcw * 0.5f, b1x2 = px + pcw * 0.5f;
    const float b1y1 = py - pch * 0.5f, b1y2 = py + pch * 0.5f;
    const float t0 = ts[0], t1 = ts[1], t2 = ts[2], t3 = ts[3];
    const float b2x1 = t0 - t2 * 0.5f, b2x2 = t0 + t2 * 0.5f;
    const float b2y1 = t1 - t3 * 0.5f, b2y2 = t1 + t3 * 0.5f;
    const float iw = fmaxf(fminf(b1x2, b2x2) - fmaxf(b1x1, b2x1), 0.f);
    const float ih = fmaxf(fminf(b1y2, b2y2) - fmaxf(b1y1, b2y1), 0.f);
    const float inter = iw * ih;
    const float w1 = b1x2 - b1x1, h1 = b1y2 - b1y1 + eps;
    const float w2 = b2x2 - b2x1, h2 = b2y2 - b2y1 + eps;
    const float uni = w1 * h1 + w2 * h2 - inter + eps;
    const float iou = inter / uni;
    const float cw = fmaxf(b1x2, b2x2) - fminf(b1x1, b2x1);
    const float chh = fmaxf(b1y2, b2y2) - fminf(b1y1, b2y1);
    const float c2 = cw * cw + chh * chh + eps;
    const float dx = b2x1 + b2x2 - b1x1 - b1x2;
    const float dy = b2y1 + b2y2 - b1y1 - b1y2;
    const float rho2 = (dx * dx + dy * dy) * 0.25f;
    const float da = atanf(w2 / h2) - atanf(w1 / h1);
    const float vv = 0.40528473456f * da * da;   // 4/pi^2
    const float alpha = vv / (vv - iou + (1.f + eps));
    const float ciou = iou - (rho2 / c2 + vv * alpha);
    t_ciou = ciou;

    // one_iou: pbox corners (b1*) vs ts interpreted as xyxy
    const float ix1 = fmaxf(b1x1, t0), iy1 = fmaxf(b1y1, t1);
    const float ix2 = fminf(b1x2, t2), iy2 = fminf(b1y2, t3);
    const float inter2 = fmaxf(ix2 - ix1, 0.f) * fmaxf(iy2 - iy1, 0.f);
    const float a1 = (b1x2 - b1x1) * (b1y2 - b1y1);
    const float a2 = (t2 - t0) * (t3 - t1);
    const float one_iou = inter2 / (a1 + a2 - inter2);

    const float o = ps[4];
    t_lobj = fmaxf(o, 0.f) - o * one_iou + log1pf(__expf(-fabsf(o)));
    t_bce0 = bce0(o);

    // class NLL: log_softmax over ps[5:85]
    int label = (int)tgt[(long)pos * 10 + 9];
    label = min(max(label, 0), 79);
    float m = -3.402823466e+38f;
#pragma unroll 8
    for (int k = 0; k < 80; ++k) m = fmaxf(m, ps[5 + k]);
    float se = 0.f;
#pragma unroll 8
    for (int k = 0; k < 80; ++k) se += __expf(ps[5 + k] - m);
    t_lcls = (m + __logf(se)) - ps[5 + label];
  }
  float r0 = block_sum256(t_ciou);
  float r1 = block_sum256(t_lobj);
  float r2 = block_sum256(t_bce0);
  float r3 = block_sum256(t_lcls);
  if (threadIdx.x == 0) {
    partials[blockIdx.x]          = r0;
    partials[GG + blockIdx.x]     = r1;
    partials[2 * GG + blockIdx.x] = r2;
    partials[3 * GG + blockIdx.x] = r3;
  }
}

// ---------------------------------------------------------------------------
// Kernel 4: combine partials -> scalar loss.
// ---------------------------------------------------------------------------
__global__ __launch_bounds__(256) void finalize_kernel(
    const float* __restrict__ pA, int GA, const float* __restrict__ pS, int GS,
    const float* __restrict__ pG, int GG, int N, int nPos, int npix,
    float* __restrict__ out) {
  float a = 0.f;
  for (int i = threadIdx.x; i < GA; i += 256) a += pA[i];
  const float sAll0 = block_sum256(a);
  float s = 0.f;
  for (int i = threadIdx.x; i < GS; i += 256) s += pS[i];
  const float sSeg = block_sum256(s);
  if (threadIdx.x == 0) {
    float sC = 0.f, sO = 0.f, sB = 0.f, sL = 0.f;
    for (int i = 0; i < GG; ++i) {
      sC += pG[i];
      sO += pG[GG + i];
      sB += pG[2 * GG + i];
      sL += pG[3 * GG + i];
    }
    const float invN = 1.f / (float)N;
    const float lbox   = 1.f - sC * invN;
    const float lobj   = sO * invN;
    const float lnoobj = (sAll0 - sB) / (float)(nPos - N);
    const float lcls   = sL * invN;
    const float lseg   = sSeg / (float)npix;
    out[0] = 3.f * lbox + 1.f * lobj + 10.f * lnoobj + lcls + 2.f * lseg;
  }
}

extern "C" void kernel_launch(void* const* d_in, const int* in_sizes, int n_in,
                              void* d_out, int out_size, void* d_ws, size_t ws_size,
                              hipStream_t stream) {
  const float* prd    = (const float*)d_in[0];
  const float* tgt    = (const float*)d_in[1];
  const int*   idx    = (const int*)d_in[2];
  const int*   fidx   = (const int*)d_in[3];
  const float* seg    = (const float*)d_in[4];
  const int*   seglab = (const int*)d_in[5];
  float* out = (float*)d_out;
  float* ws  = (float*)d_ws;

  const int A = 3, H = 80, W = 80, CH = 85;
  const int nPrd = in_sizes[0];          // 26,112,000
  const int nPos = in_sizes[1] / 10;     // 307,200
  const int N    = in_sizes[2] / 4;      // 4096
  const int npix = in_sizes[5];          // 409,600
  const int hw   = 160 * 160;

  const int GA = 1024;
  const int GS = (npix + 255) / 256;     // 1600
  const int GG = (N + 255) / 256;        // 16

  float* pA = ws;
  float* pS = pA + GA;
  float* pG = pS + GS;

  noobj_kernel<<<GA, 256, 0, stream>>>(prd, nPrd, CH, pA);
  seg_kernel<<<GS, 256, 0, stream>>>(seg, seglab, npix, hw, pS);
  gather_kernel<<<GG, 256, 0, stream>>>(prd, tgt, idx, fidx, N, A, H, W, CH, pG, GG);
  finalize_kernel<<<1, 256, 0, stream>>>(pA, GA, pS, GS, pG, GG, N, nPos, npix, out);
}